// ETCModel_53953379172978
// MI455X (gfx1250) — compile-verified
//
#include <hip/hip_runtime.h>
#include <math.h>
#include <stdint.h>

#define Bb 16
#define Ll 256
#define Hh 6
#define Dd 768
#define dh 128
#define Cc 20
#define CP 21
#define MAXF_ 20.0f
#define BN_EPS_ 1e-5f

typedef float v2f __attribute__((ext_vector_type(2)));
typedef float v8f __attribute__((ext_vector_type(8)));

// Workspace layout (floats)
#define X_OFF   0u
#define X_SZ    (Bb*Ll*Dd)              // 3,145,728
#define CO_OFF  (X_OFF + X_SZ)
#define CO_SZ   (Bb*Hh*Ll*Ll)           // 6,291,456
#define ST_OFF  (CO_OFF + CO_SZ)
#define ST_SZ   48                      // vSum[6] vSq[6] coSum[6] coSq[6] vMean[6] vRstd[6] coMean[6] coRstd[6]
#define ACC_OFF (ST_OFF + ST_SZ)
#define ACC_SZ  (Bb*CP)

__device__ __forceinline__ v8f wmma4(v2f a, v2f b, v8f c) {
  // D = A(16x4, f32) * B(4x16, f32) + C(16x16, f32)
  return __builtin_amdgcn_wmma_f32_16x16x4_f32(false, a, false, b, (short)0, c,
                                               false, false);
}

// CDNA5 async global->LDS copy (GV mode, per-lane 16B), tracked by ASYNCcnt.
// LDS byte offset = low 32 bits of the generic LDS pointer.
__device__ __forceinline__ void async_ld_b128(void* lds_ptr, const float* gaddr) {
  unsigned loff = (unsigned)(uintptr_t)lds_ptr;
  asm volatile("global_load_async_to_lds_b128 %0, %1, off"
               :: "v"(loff), "v"(gaddr)
               : "memory");
}
__device__ __forceinline__ void wait_async0() {
  asm volatile("s_wait_asynccnt 0x0" ::: "memory");
}

// ---------------- Kernel 1: scaled embedding gather + V BN partial sums ----
__global__ void k_embed(const int* __restrict__ tids,
                        const float* __restrict__ TFs,
                        const float* __restrict__ DFs,
                        const float* __restrict__ emb,
                        float* __restrict__ x, float* __restrict__ stats) {
  int bl = blockIdx.x;                       // b*L + l
  int tid = tids[bl];
  float s = fminf(TFs[bl], MAXF_) * log1pf(1.0f / DFs[bl]);
  __shared__ float hsum[Hh], hsq[Hh];
  if (threadIdx.x < Hh) { hsum[threadIdx.x] = 0.f; hsq[threadIdx.x] = 0.f; }
  __syncthreads();
  const float* er = emb + (size_t)tid * Dd;
  float* xr = x + (size_t)bl * Dd;
  for (int k = 0; k < 3; ++k) {
    int e = threadIdx.x + (k << 8);
    float v = er[e] * s;
    xr[e] = v;
    atomicAdd(&hsum[e / dh], v);
    atomicAdd(&hsq[e / dh], v * v);
  }
  __syncthreads();
  if (threadIdx.x < Hh) {
    atomicAdd(&stats[0 + threadIdx.x], hsum[threadIdx.x]);
    atomicAdd(&stats[6 + threadIdx.x], hsq[threadIdx.x]);
  }
}

// ---------------- Kernel 2: Gram matrix via WMMA -> pairwise distance ------
// One block per (b,h): 512 threads = 16 waves; X (256x128) staged in LDS
// via async global->LDS copies.
__global__ void k_gram(const float* __restrict__ x, float* __restrict__ co,
                       float* __restrict__ stats) {
  extern __shared__ float sm[];
  float* Xs  = sm;                 // 256*132
  float* sq  = sm + 256 * 132;     // 256
  float* red = sq + 256;           // 512
  int bh = blockIdx.x, b = bh / Hh, h = bh % Hh;
  int t = threadIdx.x;
  const float* xg = x + (size_t)b * Ll * Dd + h * dh;
  for (int i = t; i < 8192; i += 512) {          // 8192 float4 = 256x128
    int row = i >> 5, cg = (i & 31) << 2;
    async_ld_b128(&Xs[row * 132 + cg], xg + (size_t)row * Dd + cg);
  }
  wait_async0();                   // this wave's async copies landed in LDS
  __syncthreads();                 // publish all waves' staging
  for (int r = t; r < 256; r += 512) {
    float s = 0.f; const float* p = &Xs[r * 132];
    for (int c = 0; c < dh; ++c) s += p[c] * p[c];
    sq[r] = s;
  }
  __syncthreads();
  int wave = t >> 5, lane = t & 31, m = lane & 15, half = lane >> 4;
  float lsum = 0.f, lsq = 0.f;
  float* cobh = co + (size_t)bh * Ll * Ll;
  int it = wave;                                  // wave owns one i-tile strip
  for (int jt = 0; jt < 16; ++jt) {
    v8f acc = {0.f, 0.f, 0.f, 0.f, 0.f, 0.f, 0.f, 0.f};
    for (int k = 0; k < dh; k += 4) {
      int kk = k + 2 * half;
      v2f a, bv;
      a.x  = Xs[(it * 16 + m) * 132 + kk];
      a.y  = Xs[(it * 16 + m) * 132 + kk + 1];
      bv.x = Xs[(jt * 16 + m) * 132 + kk];        // B[k,n] = X[n,k] (n=m)
      bv.y = Xs[(jt * 16 + m) * 132 + kk + 1];
      acc = wmma4(a, bv, acc);
    }
    for (int r = 0; r < 8; ++r) {
      int i = it * 16 + r + 8 * half;
      int j = jt * 16 + m;
      float g = acc[r];
      float dist = sqrtf(fmaxf(sq[i] + sq[j] - 2.f * g, 1e-12f));
      cobh[i * Ll + j] = dist;
      lsum += dist; lsq += dist * dist;
    }
  }
  red[t] = lsum; __syncthreads();
  for (int s2 = 256; s2 > 0; s2 >>= 1) { if (t < s2) red[t] += red[t + s2]; __syncthreads(); }
  if (t == 0) atomicAdd(&stats[12 + h], red[0]);
  __syncthreads();
  red[t] = lsq; __syncthreads();
  for (int s2 = 256; s2 > 0; s2 >>= 1) { if (t < s2) red[t] += red[t + s2]; __syncthreads(); }
  if (t == 0) atomicAdd(&stats[18 + h], red[0]);
}

// ---------------- Kernel 3: finalize both BN param sets --------------------
__global__ void k_finalize(float* __restrict__ stats) {
  int h = threadIdx.x;
  if (h < Hh) {
    const float nV = (float)(Bb * Ll * dh);
    float m = stats[h] / nV;
    float v = stats[6 + h] / nV - m * m;
    stats[24 + h] = m; stats[30 + h] = rsqrtf(v + BN_EPS_);
    const float nC = (float)(Bb * Ll * Ll);
    float mc = stats[12 + h] / nC;
    float vc = stats[18 + h] / nC - mc * mc;
    stats[36 + h] = mc; stats[42 + h] = rsqrtf(vc + BN_EPS_);
  }
}

// ---------------- Kernel 4: bn -> leaky9 -> mask -> row softmax (in place) -
__global__ void k_softmax(float* __restrict__ co, const int* __restrict__ tids,
                          const float* __restrict__ stats) {
  int idx = blockIdx.x;                 // (b*H + h)*L + i
  int i = idx % Ll, bh = idx / Ll, h = bh % Hh, b = bh / Hh;
  int j = threadIdx.x;
  float* row = co + (size_t)idx * Ll;
  float mean = stats[36 + h], rstd = stats[42 + h];
  bool vi = tids[b * Ll + i] != 0;
  bool vj = tids[b * Ll + j] != 0;
  float v = (row[j] - mean) * rstd;
  v = v >= 0.f ? v : 9.f * v;
  if (!(vi && vj)) v = 0.f;
  __shared__ float red[Ll];
  red[j] = v; __syncthreads();
  for (int s = 128; s > 0; s >>= 1) { if (j < s) red[j] = fmaxf(red[j], red[j + s]); __syncthreads(); }
  float mx = red[0]; __syncthreads();
  float e = expf(v - mx);
  red[j] = e; __syncthreads();
  for (int s = 128; s > 0; s >>= 1) { if (j < s) red[j] += red[j + s]; __syncthreads(); }
  row[j] = e / red[0];
}

// ---------------- Kernel 5: V_out = P @ Vn via WMMA, BN folded, in place ---
__global__ void k_av(const float* __restrict__ co, float* __restrict__ x,
                     const float* __restrict__ stats) {
  extern __shared__ float sm[];
  float* Xs = sm;                       // 256*132
  int bh = blockIdx.x, b = bh / Hh, h = bh % Hh;
  int t = threadIdx.x;
  float* xg = x + (size_t)b * Ll * Dd + h * dh;
  for (int i = t; i < 8192; i += 512) {
    int row = i >> 5, cg = (i & 31) << 2;
    async_ld_b128(&Xs[row * 132 + cg], xg + (size_t)row * Dd + cg);
  }
  wait_async0();
  __syncthreads();                      // whole slice staged before any write
  int wave = t >> 5, lane = t & 31, m = lane & 15, half = lane >> 4;
  float mean = stats[24 + h], rstd = stats[30 + h];
  const float* P = co + (size_t)bh * Ll * Ll;
  int it = wave;
  for (int nt = 0; nt < 8; ++nt) {
    v8f acc = {0.f, 0.f, 0.f, 0.f, 0.f, 0.f, 0.f, 0.f};
    for (int k = 0; k < Ll; k += 4) {
      int kk = k + 2 * half;
      v2f a, bv;
      a.x  = P[(it * 16 + m) * Ll + kk];          // A = P tile (16x4)
      a.y  = P[(it * 16 + m) * Ll + kk + 1];
      bv.x = Xs[kk * 132 + nt * 16 + m];          // B[k,n] = X[k, nt*16+n]
      bv.y = Xs[(kk + 1) * 132 + nt * 16 + m];
      acc = wmma4(a, bv, acc);
    }
    for (int r = 0; r < 8; ++r) {
      int i = it * 16 + r + 8 * half;
      int dd2 = nt * 16 + m;
      // Vn = (X-mean)*rstd ; rows of P sum to 1 => affine folds out of GEMM
      xg[(size_t)i * Dd + dd2] = acc[r] * rstd - mean * rstd;
    }
  }
}

// ---------------- Kernel 6: per-token fc + softmax(21) + accumulate over L -
__global__ void k_fc(const float* __restrict__ x, const float* __restrict__ fcW,
                     const float* __restrict__ fcb, float* __restrict__ acc) {
  __shared__ float row[Dd];
  __shared__ float y[CP];
  int bl = blockIdx.x, b = bl / Ll;
  int t = threadIdx.x;
  const float* xr = x + (size_t)bl * Dd;
  for (int k = t; k < Dd; k += 256) row[k] = xr[k];
  __syncthreads();
  if (t < CP) {
    float s = fcb[t];
    const float* w = fcW + (size_t)t * Dd;
    for (int e = 0; e < Dd; ++e) s += row[e] * w[e];
    y[t] = s;
  }
  __syncthreads();
  if (t < CP) {
    float mx = -1e30f;
    for (int c = 0; c < CP; ++c) mx = fmaxf(mx, y[c]);
    float den = 0.f;
    for (int c = 0; c < CP; ++c) den += expf(y[c] - mx);
    atomicAdd(&acc[b * CP + t], expf(y[t] - mx) / den);
  }
}

// ---------------- Kernel 7: final softmax over first 20 classes ------------
__global__ void k_out(const float* __restrict__ acc, float* __restrict__ out) {
  int b = threadIdx.x;
  if (b < Bb) {
    float mx = -1e30f;
    for (int c = 0; c < Cc; ++c) mx = fmaxf(mx, acc[b * CP + c]);
    float e[Cc]; float den = 0.f;
    for (int c = 0; c < Cc; ++c) { e[c] = expf(acc[b * CP + c] - mx); den += e[c]; }
    for (int c = 0; c < Cc; ++c) out[b * Cc + c] = e[c] / den;
  }
}

extern "C" void kernel_launch(void* const* d_in, const int* in_sizes, int n_in,
                              void* d_out, int out_size, void* d_ws, size_t ws_size,
                              hipStream_t stream) {
  (void)in_sizes; (void)n_in; (void)out_size; (void)ws_size;
  const int*   tids = (const int*)d_in[0];
  const float* TFs  = (const float*)d_in[1];
  const float* DFs  = (const float*)d_in[2];
  const float* emb  = (const float*)d_in[3];
  const float* fcW  = (const float*)d_in[4];
  const float* fcb  = (const float*)d_in[5];
  // d_in[6]=weiW, d_in[7]=weib: mathematically dead (softmax over size-1 axis == 1)
  float* ws    = (float*)d_ws;
  float* x     = ws + X_OFF;
  float* co    = ws + CO_OFF;
  float* stats = ws + ST_OFF;
  float* acc   = ws + ACC_OFF;
  float* out   = (float*)d_out;

  hipMemsetAsync(stats, 0, (ST_SZ + ACC_SZ) * sizeof(float), stream);

  k_embed<<<Bb * Ll, 256, 0, stream>>>(tids, TFs, DFs, emb, x, stats);

  size_t sh_gram = (size_t)(256 * 132 + 256 + 512) * sizeof(float);
  k_gram<<<Bb * Hh, 512, sh_gram, stream>>>(x, co, stats);

  k_finalize<<<1, 64, 0, stream>>>(stats);

  k_softmax<<<Bb * Hh * Ll, Ll, 0, stream>>>(co, tids, stats);

  size_t sh_av = (size_t)(256 * 132) * sizeof(float);
  k_av<<<Bb * Hh, 512, sh_av, stream>>>(co, x, stats);

  k_fc<<<Bb * Ll, 256, 0, stream>>>(x, fcW, fcb, acc);

  k_out<<<1, 32, 0, stream>>>(acc, out);
}